// GCN_54322746360000
// MI455X (gfx1250) — compile-verified
//
#include <hip/hip_runtime.h>
#include <hip/hip_bf16.h>

#define N_NODES 50000
#define N_EDGES 800000
#define HIDDEN 128
#define N_GRAPHS 512
#define NEG_SLOPE 0.01f

typedef __attribute__((ext_vector_type(2))) float v2f;
typedef __attribute__((ext_vector_type(8))) float v8f;

// ---------------------------------------------------------------------------
// Zero a float range
__global__ void zero_kernel(float* __restrict__ p, int n) {
    int i = blockIdx.x * 256 + threadIdx.x;
    if (i < n) p[i] = 0.0f;
}

// z0[i] = x[i]  (layer-0 scalar init)
__global__ void init_z0_kernel(const float* __restrict__ x, float* __restrict__ z0) {
    int i = blockIdx.x * 256 + threadIdx.x;
    if (i < N_NODES) z0[i] = x[i];
}

// scalar edge scatter: z0[dst] += x[src]
__global__ void scatter1_kernel(const float* __restrict__ x, const int* __restrict__ ei,
                                float* __restrict__ z0) {
    int e = blockIdx.x * 256 + threadIdx.x;
    if (e < N_EDGES) {
        int src = ei[e];
        int dst = ei[N_EDGES + e];
        atomicAdd(&z0[dst], x[src]);
    }
}

// layer-0 GEMM1 equivalent: t[i][n] = leaky(z0[i]*W1[n] + b1[n])   (W1 is 1x128)
__global__ void expand0_kernel(const float* __restrict__ z0, const float* __restrict__ W1,
                               const float* __restrict__ b1, float* __restrict__ t) {
    int id = blockIdx.x * 256 + threadIdx.x;   // i*128 + n
    if (id < N_NODES * HIDDEN) {
        int i = id >> 7, n = id & 127;
        float v = z0[i] * W1[n] + b1[n];
        t[id] = v >= 0.0f ? v : NEG_SLOPE * v;
    }
}

// z[i][:] = h[i][:]   (float4 vectorized)
__global__ void copy128_kernel(const float4* __restrict__ h, float4* __restrict__ z) {
    int i = blockIdx.x * 256 + threadIdx.x;
    if (i < N_NODES * (HIDDEN / 4)) z[i] = h[i];
}

// wave-per-edge 128-wide scatter: z[dst][:] += h[src][:]
__global__ void scatter128_kernel(const float* __restrict__ h, const int* __restrict__ ei,
                                  float* __restrict__ z) {
    int wave = threadIdx.x >> 5;
    int lane = threadIdx.x & 31;
    int e = blockIdx.x * 8 + wave;
    if (e >= N_EDGES) return;
    int src = ei[e];
    int dst = ei[N_EDGES + e];
    float4 v = *(const float4*)(h + (size_t)src * HIDDEN + lane * 4);
    float* d = z + (size_t)dst * HIDDEN + lane * 4;
    atomicAdd(d + 0, v.x);
    atomicAdd(d + 1, v.y);
    atomicAdd(d + 2, v.z);
    atomicAdd(d + 3, v.w);
}

// ---------------------------------------------------------------------------
// WMMA fp32 GEMM: out[M x 128] = act(A[M x 128] @ W[128 x 128] + bias)
// act: 0=none, 1=leaky, 2=relu
// 256 threads = 8 waves; wave owns 16 rows, full 128 cols (8 C tiles).
// W staged in LDS in per-fragment float2 layout: fragment (ks,nt), lane l holds
// (W[4ks+2*(l>>4)][16nt+(l&15)], W[4ks+2*(l>>4)+1][16nt+(l&15)]) as one b64.
__global__ __launch_bounds__(256) void gemm128_kernel(const float* __restrict__ A,
                                                      const float* __restrict__ W,
                                                      const float* __restrict__ bias,
                                                      float* __restrict__ out,
                                                      int M, int act) {
    __shared__ float Wlds[HIDDEN * HIDDEN];
    const int tid = threadIdx.x;
    // cooperative stage + swizzle of W (row-major [k][n]) into fragment layout
    for (int i = 0; i < 64; ++i) {
        int id = tid + (i << 8);               // id = k*128 + n
        int k = id >> 7, n = id & 127;
        float w = W[id];
        int ks = k >> 2, r = k & 3;
        int kh = r >> 1, half = r & 1;
        int lane_t = (kh << 4) | (n & 15);
        int nt = n >> 4;
        Wlds[((((ks << 3) | nt) << 5) | lane_t) * 2 + half] = w;
    }
    __syncthreads();

    const int wave = tid >> 5, lane = tid & 31;
    const int row0 = blockIdx.x * 128 + wave * 16;
    if (row0 + 16 > M) return;                  // M % 16 == 0 -> uniform per wave
    const int m16 = lane & 15, kh = lane >> 4;

    v8f acc[8];
#pragma unroll
    for (int nt = 0; nt < 8; ++nt) {
        float bv = bias[nt * 16 + m16];         // bias depends only on column
#pragma unroll
        for (int j = 0; j < 8; ++j) acc[nt][j] = bv;
    }

    const float* Arow = A + (size_t)(row0 + m16) * HIDDEN + 2 * kh;
#pragma unroll 4
    for (int ks = 0; ks < 32; ++ks) {
        v2f a = *(const v2f*)(Arow + ks * 4);   // A frag: K = 4ks+2kh, +1
#pragma unroll
        for (int nt = 0; nt < 8; ++nt) {
            v2f b = *(const v2f*)(&Wlds[((((ks << 3) | nt) << 5) | lane) * 2]);
            acc[nt] = __builtin_amdgcn_wmma_f32_16x16x4_f32(
                false, a, false, b, (short)0, acc[nt], false, false);
        }
    }

    // epilogue: C tile layout -> lanes 0-15 row v, lanes 16-31 row v+8
#pragma unroll
    for (int nt = 0; nt < 8; ++nt) {
#pragma unroll
        for (int v = 0; v < 8; ++v) {
            int m = v + (kh << 3);
            float val = acc[nt][v];
            if (act == 1) val = val >= 0.0f ? val : NEG_SLOPE * val;
            else if (act == 2) val = val > 0.0f ? val : 0.0f;
            out[(size_t)(row0 + m) * HIDDEN + nt * 16 + m16] = val;
        }
    }
}

// ---------------------------------------------------------------------------
// wave-per-node pooling: sums[batch[i]][:] += h[i][:], counts[batch[i]] += 1
__global__ void pool_kernel(const float* __restrict__ h, const int* __restrict__ batch,
                            float* __restrict__ sums, float* __restrict__ counts) {
    int wave = threadIdx.x >> 5;
    int lane = threadIdx.x & 31;
    int node = blockIdx.x * 8 + wave;
    if (node >= N_NODES) return;
    int g = batch[node];
    float4 v = *(const float4*)(h + (size_t)node * HIDDEN + lane * 4);
    float* d = sums + (size_t)g * HIDDEN + lane * 4;
    atomicAdd(d + 0, v.x);
    atomicAdd(d + 1, v.y);
    atomicAdd(d + 2, v.z);
    atomicAdd(d + 3, v.w);
    if (lane == 0) atomicAdd(&counts[g], 1.0f);
}

// per-graph head: pooled = sums/max(count,1); logits = pooled@linW + linb; log_softmax
__global__ void head_kernel(const float* __restrict__ sums, const float* __restrict__ counts,
                            const float* __restrict__ linW, const float* __restrict__ linb,
                            float* __restrict__ out) {
    int g = blockIdx.x * 256 + threadIdx.x;
    if (g >= N_GRAPHS) return;
    float c = counts[g];
    float inv = 1.0f / fmaxf(c, 1.0f);
    float l0 = 0.0f, l1 = 0.0f;
    const float* s = sums + (size_t)g * HIDDEN;
    for (int n = 0; n < HIDDEN; ++n) {
        float p = s[n] * inv;
        l0 += p * linW[n * 2 + 0];
        l1 += p * linW[n * 2 + 1];
    }
    l0 += linb[0];
    l1 += linb[1];
    float m = fmaxf(l0, l1);
    float lse = m + logf(expf(l0 - m) + expf(l1 - m));
    out[g * 2 + 0] = l0 - lse;
    out[g * 2 + 1] = l1 - lse;
}

// ---------------------------------------------------------------------------
extern "C" void kernel_launch(void* const* d_in, const int* in_sizes, int n_in,
                              void* d_out, int out_size, void* d_ws, size_t ws_size,
                              hipStream_t stream) {
    (void)in_sizes; (void)n_in; (void)out_size; (void)ws_size;
    // input order: x, 4x(W1,b1,W2,b2), lin_W, lin_b, edge_index, batch
    const float* x = (const float*)d_in[0];
    const float* W1[4]; const float* b1[4]; const float* W2[4]; const float* b2[4];
    for (int l = 0; l < 4; ++l) {
        W1[l] = (const float*)d_in[1 + 4 * l];
        b1[l] = (const float*)d_in[2 + 4 * l];
        W2[l] = (const float*)d_in[3 + 4 * l];
        b2[l] = (const float*)d_in[4 + 4 * l];
    }
    const float* linW  = (const float*)d_in[17];
    const float* linb  = (const float*)d_in[18];
    const int*   ei    = (const int*)d_in[19];
    const int*   batch = (const int*)d_in[20];

    float* ws = (float*)d_ws;
    size_t off = 0;
    float* z0   = ws + off; off += 50048;                 // [N] scalar z for layer 0
    float* zbuf = ws + off; off += (size_t)N_NODES * HIDDEN;  // z = h + agg
    float* tbuf = ws + off; off += (size_t)N_NODES * HIDDEN;  // MLP hidden
    float* hbuf = ws + off; off += (size_t)N_NODES * HIDDEN;  // node features
    float* sums = ws + off; off += (size_t)N_GRAPHS * HIDDEN; // pooled sums
    float* cnts = ws + off; off += N_GRAPHS;                  // pooled counts

    const int gemmBlocks  = (N_NODES + 127) / 128;        // 391
    const int edgeWaveBlk = (N_EDGES + 7) / 8;            // 100000
    const int nodeWaveBlk = (N_NODES + 7) / 8;            // 6250
    const int copyBlocks  = (N_NODES * (HIDDEN / 4) + 255) / 256;

    // zero pooled sums + counts (contiguous)
    zero_kernel<<<(N_GRAPHS * HIDDEN + N_GRAPHS + 255) / 256, 256, 0, stream>>>(
        sums, N_GRAPHS * HIDDEN + N_GRAPHS);

    // ---- layer 0 (d_in = 1) ----
    init_z0_kernel<<<(N_NODES + 255) / 256, 256, 0, stream>>>(x, z0);
    scatter1_kernel<<<(N_EDGES + 255) / 256, 256, 0, stream>>>(x, ei, z0);
    expand0_kernel<<<(N_NODES * HIDDEN + 255) / 256, 256, 0, stream>>>(z0, W1[0], b1[0], tbuf);
    gemm128_kernel<<<gemmBlocks, 256, 0, stream>>>(tbuf, W2[0], b2[0], hbuf, N_NODES, 1);

    // ---- layers 1..3 ----
    for (int l = 1; l < 4; ++l) {
        copy128_kernel<<<copyBlocks, 256, 0, stream>>>((const float4*)hbuf, (float4*)zbuf);
        scatter128_kernel<<<edgeWaveBlk, 256, 0, stream>>>(hbuf, ei, zbuf);
        gemm128_kernel<<<gemmBlocks, 256, 0, stream>>>(zbuf, W1[l], b1[l], tbuf, N_NODES, 1);
        int act2 = (l < 3) ? 1 : 2;  // inter-layer leaky, final relu for pooling
        gemm128_kernel<<<gemmBlocks, 256, 0, stream>>>(tbuf, W2[l], b2[l], hbuf, N_NODES, act2);
    }

    // ---- global mean pool + linear + log_softmax ----
    pool_kernel<<<nodeWaveBlk, 256, 0, stream>>>(hbuf, batch, sums, cnts);
    head_kernel<<<(N_GRAPHS + 255) / 256, 256, 0, stream>>>(sums, cnts, linW, linb, (float*)d_out);
}